// ReconStep_78348793414179
// MI455X (gfx1250) — compile-verified
//
#include <hip/hip_runtime.h>

// CDNA5 (gfx1250) wave32 WMMA: V_WMMA_F32_16X16X4_F32, A/B = 2 VGPRs, C/D = 8.
typedef float v2f __attribute__((ext_vector_type(2)));
typedef float v8f __attribute__((ext_vector_type(8)));

#define MLEM_EPS 1e-8f

static __device__ __forceinline__ v8f wmma4f32(v2f a, v2f b, v8f c) {
  // 8 args: (neg_a, A, neg_b, B, c_mod, C, reuse_a, reuse_b)
  return __builtin_amdgcn_wmma_f32_16x16x4_f32(false, a, false, b,
                                               (short)0, c, false, false);
}

// ---------------------------------------------------------------------------
// Forward + ratio:  ratio[d,c] = sinos[d,c] / (sum_v M[d,v]*img[v,c] + EPS)
// Oriented as D(c, n) = A(c,k) * B(k,n):  A = img^T k-slab, B = M tile.
// Lanes m>=4 load in-bounds garbage into A rows 4..15 -> only dead D rows are
// polluted (we read D rows 0..3 only), so the hot loop has NO predication.
// Each wave: TWO 16-row det tiles, waves split v; LDS cross-wave reduction.
// ---------------------------------------------------------------------------
__global__ __launch_bounds__(256) void mlem_fwd_ratio(
    const float* __restrict__ M, const float* __restrict__ img,
    const float* __restrict__ sinos, float* __restrict__ ratio,
    int n_det, int n_vox) {
  const int lane = threadIdx.x & 31;
  const int wave = threadIdx.x >> 5;      // 0..7
  const int m    = lane & 15;
  const int half = lane >> 4;             // selects K pair {0,1} vs {2,3}
  const int d0   = blockIdx.x * 32;       // two 16-row det tiles

  const int vslice = n_vox >> 3;          // per-wave reduction slice
  const int vbeg   = wave * vslice;
  const int nit    = vslice >> 2;         // k-steps of 4

  // B: float2 per lane, contiguous within a row -> global_load_b64.
  const float* bp0 = M + (size_t)(d0 + m) * (size_t)n_vox + (size_t)(2 * half + vbeg);
  const float* bp1 = bp0 + (size_t)16 * (size_t)n_vox;
  // A: img[v0+k][c] with clamped c = m&3 (garbage rows 4..15 are dead).
  const float* ap  = img + (size_t)(vbeg + 2 * half) * 4 + (m & 3);

  v8f acc0 = {}, acc1 = {};
  #pragma unroll 4
  for (int it = 0; it < nit; ++it) {
    v2f b0 = *(const v2f*)bp0;
    v2f b1 = *(const v2f*)bp1;
    v2f a;
    a.x = ap[0];                          // img[v0+2*half    ][m&3]
    a.y = ap[4];                          // img[v0+2*half + 1][m&3]
    acc0 = wmma4f32(a, b0, acc0);
    acc1 = wmma4f32(a, b1, acc1);
    bp0 += 4; bp1 += 4; ap += 16;
  }

  // D: VGPR r, lanes 0-15 hold D[r, n] = proj[d0+n][c=r] (rows 0..3 valid).
  __shared__ float red[8][32][4];
  if (lane < 16) {
    #pragma unroll
    for (int c = 0; c < 4; ++c) {
      red[wave][lane][c]      = acc0[c];
      red[wave][lane + 16][c] = acc1[c];
    }
  }
  __syncthreads();

  if (threadIdx.x < 32) {
    const int n = threadIdx.x;
    float4 p = make_float4(0.f, 0.f, 0.f, 0.f);
    #pragma unroll
    for (int w = 0; w < 8; ++w) {
      p.x += red[w][n][0]; p.y += red[w][n][1];
      p.z += red[w][n][2]; p.w += red[w][n][3];
    }
    const float4 s = *(const float4*)(sinos + (size_t)(d0 + n) * 4);
    float4 r;
    r.x = s.x / (p.x + MLEM_EPS);
    r.y = s.y / (p.y + MLEM_EPS);
    r.z = s.z / (p.z + MLEM_EPS);
    r.w = s.w / (p.w + MLEM_EPS);
    *(float4*)(ratio + (size_t)(d0 + n) * 4) = r;
  }
}

// ---------------------------------------------------------------------------
// Backward + update: out[v,c] = img[v,c]*effmap[v,c]* sum_d M[d,v]*ratio[d,c]
// D(c, n) with n over vox; A = ratio^T k-slab, B[k,n] = M[d0+k][v0+n].
// Two adjacent 16-vox tiles per wave -> the four row-segment loads per k-step
// cover full 128B lines. Waves split d; LDS cross-wave reduction.
// ---------------------------------------------------------------------------
__global__ __launch_bounds__(256) void mlem_bwd_update(
    const float* __restrict__ M, const float* __restrict__ ratio,
    const float* __restrict__ img, const float* __restrict__ effmap,
    float* __restrict__ out, int n_det, int n_vox) {
  const int lane = threadIdx.x & 31;
  const int wave = threadIdx.x >> 5;
  const int m    = lane & 15;
  const int half = lane >> 4;
  const int v0   = blockIdx.x * 32;       // two 16-wide vox tiles

  const int dslice = n_det >> 3;
  const int dbeg   = wave * dslice;
  const int nit    = dslice >> 2;         // k-steps of 4

  // A: ratio[d0+k][c] with clamped c = m&3 (rows 4..15 of D are dead).
  const float* ap = ratio + (size_t)(dbeg + 2 * half) * 4 + (m & 3);
  // B: per-lane dword at row d0+2*half(+1), col v0+m (and +16 for tile 1).
  const float* pb = M + (size_t)(dbeg + 2 * half) * (size_t)n_vox + (size_t)(v0 + m);

  v8f acc0 = {}, acc1 = {};
  #pragma unroll 4
  for (int it = 0; it < nit; ++it) {
    v2f a;
    a.x = ap[0];                          // ratio[d0+2*half    ][m&3]
    a.y = ap[4];                          // ratio[d0+2*half + 1][m&3]
    v2f b0, b1;
    b0.x = pb[0];                         // M[d0+2*half    ][v0+m]
    b0.y = pb[n_vox];                     // M[d0+2*half + 1][v0+m]
    b1.x = pb[16];                        // second tile
    b1.y = pb[n_vox + 16];
    acc0 = wmma4f32(a, b0, acc0);
    acc1 = wmma4f32(a, b1, acc1);
    ap += 16; pb += (size_t)4 * (size_t)n_vox;
  }

  // D: VGPR r, lanes 0-15 -> bp[v0+n][c=r] (rows 0..3 valid).
  __shared__ float red[8][32][4];
  if (lane < 16) {
    #pragma unroll
    for (int c = 0; c < 4; ++c) {
      red[wave][lane][c]      = acc0[c];
      red[wave][lane + 16][c] = acc1[c];
    }
  }
  __syncthreads();

  if (threadIdx.x < 32) {
    const int n = threadIdx.x;
    float4 bp4 = make_float4(0.f, 0.f, 0.f, 0.f);
    #pragma unroll
    for (int w = 0; w < 8; ++w) {
      bp4.x += red[w][n][0]; bp4.y += red[w][n][1];
      bp4.z += red[w][n][2]; bp4.w += red[w][n][3];
    }
    const size_t idx = (size_t)(v0 + n) * 4;
    const float4 iv = *(const float4*)(img + idx);
    const float4 ev = *(const float4*)(effmap + idx);
    float4 o;
    o.x = iv.x * ev.x * bp4.x;
    o.y = iv.y * ev.y * bp4.y;
    o.z = iv.z * ev.z * bp4.z;
    o.w = iv.w * ev.w * bp4.w;
    *(float4*)(out + idx) = o;
  }
}

extern "C" void kernel_launch(void* const* d_in, const int* in_sizes, int n_in,
                              void* d_out, int out_size, void* d_ws, size_t ws_size,
                              hipStream_t stream) {
  const float* img     = (const float*)d_in[0];
  const float* effmap  = (const float*)d_in[1];
  const float* sinos   = (const float*)d_in[2];
  const float* matrixs = (const float*)d_in[3];
  float* out = (float*)d_out;

  const int n_vox = in_sizes[0] / 4;   // img is [n_vox, 4]
  const int n_det = in_sizes[2] / 4;   // sinos is [n_det, 4]

  float* ratio = (float*)d_ws;         // [n_det, 4] scratch

  mlem_fwd_ratio<<<n_det / 32, 256, 0, stream>>>(matrixs, img, sinos, ratio,
                                                 n_det, n_vox);
  mlem_bwd_update<<<n_vox / 32, 256, 0, stream>>>(matrixs, ratio, img, effmap,
                                                  out, n_det, n_vox);
}